// VPNNetwork_7481833030361
// MI455X (gfx1250) — compile-verified
//
#include <hip/hip_runtime.h>

// VPN (value-propagation network) fused kernel for gfx1250 (MI455X).
// fp32 end-to-end; Phi GEMM runs on V_WMMA_F32_16X16X4_F32.
// Stage-1 obs staging uses GLOBAL_LOAD_ASYNC_TO_LDS (ASYNCcnt) when the
// toolchain exposes the builtin, else falls back to load+ds_store.

typedef __attribute__((ext_vector_type(2))) float v2f;
typedef __attribute__((ext_vector_type(4))) float v4f;
typedef __attribute__((ext_vector_type(8))) float v8f;

#define HH 4
#define WW 4
#define CC 3
#define KITER 20
#define NACT 4

#define THREADS 128            // 4 waves (wave32)
#define WAVES 4
#define SPW 32                 // samples per wave (lane == sample in VI phase)
#define SPB (WAVES * SPW)      // 128 samples per block
#define OBS_STRIDE 52          // 48 floats + pad (keeps 16B alignment, spreads banks)
#define PHI_STRIDE 52
#define V_STRIDE 17

#if __has_builtin(__builtin_amdgcn_global_load_async_to_lds_b128)
#define HAVE_ASYNC_LDS 1
#else
#define HAVE_ASYNC_LDS 0
#endif

#if HAVE_ASYNC_LDS
// Builtin signature (from hipcc diagnostic): param 1 is
//   'int __attribute__((vector_size(16))) __device__ *'  (global int4*)
// so cast to exactly that; param 2 is the LDS-side int4*.
typedef int v4i_gcc __attribute__((vector_size(16)));

__device__ static inline v4i_gcc __attribute__((address_space(1)))* to_gbl(const void* p) {
    // generic global pointer -> AS1 pointer (same 64-bit value)
    return (v4i_gcc __attribute__((address_space(1)))*)(uintptr_t)p;
}
__device__ static inline v4i_gcc __attribute__((address_space(3)))* to_lds(void* p) {
    // Generic shared-pointer low 32 bits == wave-relative LDS byte offset
    // (ISA 10.2: LDS aperture maps by truncating to addr[31:0]).
    return (v4i_gcc __attribute__((address_space(3)))*)(uint32_t)(uintptr_t)p;
}
#endif

__global__ __launch_bounds__(THREADS)
void vpn_fused_kernel(const float* __restrict__ obs,      // [B,48]
                      const float* __restrict__ Phi_w,    // [48,48]
                      const float* __restrict__ Phi_b,    // [48]
                      const float* __restrict__ Lw,       // [4,36]
                      const float* __restrict__ Lb,       // [4]
                      float* __restrict__ out)            // [B,4]
{
    __shared__ float lds_obs[SPB * OBS_STRIDE];   // 26624 B
    __shared__ float lds_phi[SPB * PHI_STRIDE];   // 26624 B
    __shared__ float lds_v  [SPB * V_STRIDE];     //  8704 B   (total 61952 B)

    const int lane = threadIdx.x & 31;
    const int wave = threadIdx.x >> 5;
    const int wave_s0 = blockIdx.x * SPB + wave * SPW;   // first global sample of this wave

    float* obs_w = lds_obs + wave * SPW * OBS_STRIDE;
    float* phi_w = lds_phi + wave * SPW * PHI_STRIDE;
    float* v_w   = lds_v   + wave * SPW * V_STRIDE;

    // ---------------- Stage 1: obs -> LDS (32 samples x 48 f32, coalesced) ------
    // 384 float4 groups; group g -> row g/12, col (g%12)*4 (48 % 4 == 0, no row crossing)
#if HAVE_ASYNC_LDS
    {
        const float* src = obs + (size_t)wave_s0 * 48;
        #pragma unroll
        for (int i = 0; i < 12; ++i) {
            int g   = i * 32 + lane;
            int row = g / 12;
            int col = (g % 12) * 4;
            __builtin_amdgcn_global_load_async_to_lds_b128(
                to_gbl(src + g * 4),
                to_lds(obs_w + row * OBS_STRIDE + col),
                /*offset=*/0, /*cpol=*/0);
        }
    }
#else
    {
        const float* src = obs + (size_t)wave_s0 * 48;
        #pragma unroll
        for (int i = 0; i < 12; ++i) {
            int g   = i * 32 + lane;
            v4f d   = *(const v4f*)(src + g * 4);
            int row = g / 12;
            int col = (g % 12) * 4;
            *(v4f*)(obs_w + row * OBS_STRIDE + col) = d;
        }
    }
#endif
    // All LDS traffic below is wave-private; DS ops are in-order within a wave,
    // so no workgroup barrier is required anywhere in this kernel.

    // ---------------- Stage 2: phi = obs @ Phi_w^T + Phi_b via WMMA f32 ----------
    // A (16x4 f32): lanes 0-15 -> M=lane, K={0,1}; lanes 16-31 -> M=lane-16, K={2,3}
    // B (4x16 f32): mirrored split; C/D (16x16 f32): lanes 0-15 N=lane M=r, 16-31 M=r+8
    const int nlo = lane & 15;
    const int khi = (lane >> 4) * 2;        // 0 or 2
    const int mhi = (lane >> 4) * 8;        // 0 or 8

    // B fragments stream from global (tiny, L2/L0-resident) while the async DMA
    // fills LDS with this wave's obs tile.
    v2f bfrag[3][12];
    #pragma unroll
    for (int nt = 0; nt < 3; ++nt)
        #pragma unroll
        for (int kc = 0; kc < 12; ++kc)
            bfrag[nt][kc] = *(const v2f*)(Phi_w + (nt * 16 + nlo) * 48 + kc * 4 + khi);

#if HAVE_ASYNC_LDS
    // Async copies must land before the A-fragment LDS reads.
#if __has_builtin(__builtin_amdgcn_s_wait_asynccnt)
    __builtin_amdgcn_s_wait_asynccnt(0);
#else
    asm volatile("s_wait_asynccnt 0x0" ::: "memory");
#endif
    asm volatile("" ::: "memory");   // keep LDS reads below the wait
#endif

    #pragma unroll
    for (int mt = 0; mt < 2; ++mt) {
        v2f afrag[12];
        #pragma unroll
        for (int kc = 0; kc < 12; ++kc)
            afrag[kc] = *(const v2f*)(obs_w + (mt * 16 + nlo) * OBS_STRIDE + kc * 4 + khi);

        #pragma unroll
        for (int nt = 0; nt < 3; ++nt) {
            float bias = Phi_b[nt * 16 + nlo];
            v8f c = {bias, bias, bias, bias, bias, bias, bias, bias};
            #pragma unroll
            for (int kc = 0; kc < 12; ++kc)
                c = __builtin_amdgcn_wmma_f32_16x16x4_f32(
                        false, afrag[kc], false, bfrag[nt][kc],
                        (short)0, c, false, false);
            // scatter tile into sample-major phi (this is the transpose)
            #pragma unroll
            for (int r = 0; r < 8; ++r) {
                int row = mt * 16 + r + mhi;
                phi_w[row * PHI_STRIDE + nt * 16 + nlo] = c[r];
            }
        }
    }

    // ---------------- Stage 3: value iteration, lane == sample, all in regs ------
    float ph[48];
    #pragma unroll
    for (int i = 0; i < 12; ++i)
        *(v4f*)(ph + i * 4) = *(const v4f*)(phi_w + lane * PHI_STRIDE + i * 4);

    float rin[16], rout[16], p[16];
    #pragma unroll
    for (int c = 0; c < 16; ++c) {
        rin[c]  = ph[c * 3 + 0];
        rout[c] = ph[c * 3 + 1];
        p[c]    = ph[c * 3 + 2];
    }

    // per directed edge: a[c][d] = rin[nbr] - (rin[nbr] != 0 ? rout[c] : 0)
    float aterm[16][4];
    #pragma unroll
    for (int c = 0; c < 16; ++c) {
        const int i = c >> 2, j = c & 3;
        const int nn[4] = { i > 0 ? c - 4 : -1, i < 3 ? c + 4 : -1,
                            j > 0 ? c - 1 : -1, j < 3 ? c + 1 : -1 };
        #pragma unroll
        for (int d = 0; d < 4; ++d) {
            if (nn[d] >= 0) {
                float rn = rin[nn[d]];
                aterm[c][d] = rn - (rn != 0.0f ? rout[c] : 0.0f);
            } else {
                aterm[c][d] = 0.0f; // unused (edge skipped below)
            }
        }
    }

    float v[16];
    #pragma unroll
    for (int c = 0; c < 16; ++c) v[c] = 0.0f;

    // OOB taps contribute exactly 0 and v >= 0 monotonically -> safe to skip them.
    for (int it = 0; it < KITER; ++it) {
        float nv[16];
        #pragma unroll
        for (int c = 0; c < 16; ++c) {
            const int i = c >> 2, j = c & 3;
            const int nn[4] = { i > 0 ? c - 4 : -1, i < 3 ? c + 4 : -1,
                                j > 0 ? c - 1 : -1, j < 3 ? c + 1 : -1 };
            float best = v[c];
            #pragma unroll
            for (int d = 0; d < 4; ++d)
                if (nn[d] >= 0)
                    best = fmaxf(best, fmaf(p[c], v[nn[d]], aterm[c][d]));
            nv[c] = best;
        }
        #pragma unroll
        for (int c = 0; c < 16; ++c) v[c] = nv[c];
    }

    #pragma unroll
    for (int c = 0; c < 16; ++c) v_w[lane * V_STRIDE + c] = v[c];

    // ---------------- Stage 4: agent position + 3x3 window gather ---------------
    const float* obs_row = obs_w + lane * OBS_STRIDE;
    int pos = 0;
    #pragma unroll
    for (int c = 15; c >= 0; --c)                 // first nonzero of one-hot ch1
        if (obs_row[c * 3 + 1] != 0.0f) pos = c;
    const int pi = pos >> 2, pj = pos & 3;

    float sel[36];
    #pragma unroll
    for (int aa = 0; aa < 3; ++aa) {
        #pragma unroll
        for (int bb = 0; bb < 3; ++bb) {
            int ii = pi - 1 + aa, jj = pj - 1 + bb;
            bool inb = (ii >= 0) && (ii < HH) && (jj >= 0) && (jj < WW);
            int cell = inb ? (ii * WW + jj) : 0;   // clamped; value masked below
            int t = (aa * 3 + bb) * 4;
            sel[t + 0] = inb ? obs_row[cell * 3 + 0] : 0.0f;
            sel[t + 1] = inb ? obs_row[cell * 3 + 1] : 0.0f;
            sel[t + 2] = inb ? obs_row[cell * 3 + 2] : 0.0f;
            sel[t + 3] = inb ? v_w[lane * V_STRIDE + cell] : 0.0f;
        }
    }

    // ---------------- Stage 5: logits = sel @ Lw^T + Lb, coalesced store --------
    float acc[NACT];
    #pragma unroll
    for (int n = 0; n < NACT; ++n) acc[n] = Lb[n];
    #pragma unroll
    for (int t = 0; t < 36; ++t) {
        float s = sel[t];
        #pragma unroll
        for (int n = 0; n < NACT; ++n)
            acc[n] = fmaf(s, Lw[n * 36 + t], acc[n]);
    }
    *(v4f*)(out + (size_t)(wave_s0 + lane) * NACT) = (v4f){acc[0], acc[1], acc[2], acc[3]};
}

extern "C" void kernel_launch(void* const* d_in, const int* in_sizes, int n_in,
                              void* d_out, int out_size, void* d_ws, size_t ws_size,
                              hipStream_t stream) {
    (void)n_in; (void)out_size; (void)d_ws; (void)ws_size;
    const float* obs   = (const float*)d_in[0];
    const float* Phi_w = (const float*)d_in[1];
    const float* Phi_b = (const float*)d_in[2];
    const float* Lw    = (const float*)d_in[3];
    const float* Lb    = (const float*)d_in[4];
    float* out = (float*)d_out;

    const int b = in_sizes[0] / 48;          // B = 262144
    const int grid = b / SPB;                // 2048 blocks of 128 threads
    vpn_fused_kernel<<<grid, THREADS, 0, stream>>>(obs, Phi_w, Phi_b, Lw, Lb, out);
}